// GNNlayer_16707422781831
// MI455X (gfx1250) — compile-verified
//
#include <hip/hip_runtime.h>
#include <hip/hip_bf16.h>

typedef __attribute__((ext_vector_type(2))) float v2f;
typedef __attribute__((ext_vector_type(8))) float v8f;

#define IN_DIM 64
#define OUT_DIM 64
#define LDS_STRIDE 68   // pad 64->68 words: B-frag ds_load_b64 hits 64 distinct banks

// ---------------- Phase 0: zero the aggregation buffer ----------------
__global__ void zero_kernel(float4* __restrict__ p, int n4) {
    int i = blockIdx.x * blockDim.x + threadIdx.x;
    if (i < n4) p[i] = make_float4(0.f, 0.f, 0.f, 0.f);
}

// ---------------- Phase 1: edge scatter-add (copy_u + sum) ----------------
// One wave32 per edge: lane l loads feat[src][2l..2l+1] (coalesced b64) and
// atomically adds into agg[dst][2l..2l+1]. unsafeAtomicAdd -> hardware
// global_atomic_add_f32 without return (STOREcnt), L2-resident target.
__global__ void scatter_kernel(const float* __restrict__ feat,
                               const int* __restrict__ src,
                               const int* __restrict__ dst,
                               float* __restrict__ agg,
                               int n_edges) {
    int e = blockIdx.x * 8 + (threadIdx.x >> 5);
    if (e >= n_edges) return;
    int lane = threadIdx.x & 31;
    int s = src[e];
    int d = dst[e];
    const float2 v = *(const float2*)(feat + (size_t)s * IN_DIM + 2 * lane);
    float* ap = agg + (size_t)d * IN_DIM + 2 * lane;
    unsafeAtomicAdd(ap, v.x);
    unsafeAtomicAdd(ap + 1, v.y);
}

// ---------------- Phase 2: out = agg @ W^T + b via V_WMMA_F32_16X16X4_F32 ----
// 8 waves/block, one 16-row M-tile per wave, all 64 output dims (4 N-tiles)
// accumulated in registers. W staged once per block into padded LDS.
__global__ void gemm_kernel(const float* __restrict__ agg,
                            const float* __restrict__ W,
                            const float* __restrict__ bias,
                            float* __restrict__ out,
                            int n_nodes) {
    __shared__ float lW[OUT_DIM * LDS_STRIDE];

    const int tid = threadIdx.x;
    // Cooperative stage of W[64][64] into LDS with padded stride.
    for (int idx = tid; idx < OUT_DIM * IN_DIM; idx += 256) {
        lW[(idx >> 6) * LDS_STRIDE + (idx & 63)] = W[idx];
    }
    __syncthreads();

    const int wave = tid >> 5;
    const int lane = tid & 31;
    const int m_tile = blockIdx.x * 8 + wave;
    const int m_tiles = (n_nodes + 15) >> 4;
    if (m_tile >= m_tiles) return;           // wave-uniform: EXEC stays all-ones

    const int half = lane >> 4;              // selects K pair within the fragment
    const int l16  = lane & 15;
    const int row  = m_tile * 16 + l16;
    const int rowc = row < n_nodes ? row : n_nodes - 1;   // clamp, keep EXEC full
    const float* arow = agg + (size_t)rowc * IN_DIM + 2 * half;

    v8f acc0 = {}, acc1 = {}, acc2 = {}, acc3 = {};

#pragma unroll
    for (int k0 = 0; k0 < IN_DIM; k0 += 4) {
        // A fragment: 16x4 f32. lane<16: K=k0,k0+1 ; lane>=16: K=k0+2,k0+3
        const float2 av = *(const float2*)(arow + k0);
        v2f a; a.x = av.x; a.y = av.y;

        const int kk = k0 + 2 * half;
        // B fragment nt: B[k][n] = W[n][k]; lane holds col n = nt*16 + l16
        const float2 bv0 = *(const float2*)(&lW[(0 * 16 + l16) * LDS_STRIDE + kk]);
        const float2 bv1 = *(const float2*)(&lW[(1 * 16 + l16) * LDS_STRIDE + kk]);
        const float2 bv2 = *(const float2*)(&lW[(2 * 16 + l16) * LDS_STRIDE + kk]);
        const float2 bv3 = *(const float2*)(&lW[(3 * 16 + l16) * LDS_STRIDE + kk]);
        v2f b0; b0.x = bv0.x; b0.y = bv0.y;
        v2f b1; b1.x = bv1.x; b1.y = bv1.y;
        v2f b2; b2.x = bv2.x; b2.y = bv2.y;
        v2f b3; b3.x = bv3.x; b3.y = bv3.y;

        acc0 = __builtin_amdgcn_wmma_f32_16x16x4_f32(false, a, false, b0, (short)0, acc0, false, false);
        acc1 = __builtin_amdgcn_wmma_f32_16x16x4_f32(false, a, false, b1, (short)0, acc1, false, false);
        acc2 = __builtin_amdgcn_wmma_f32_16x16x4_f32(false, a, false, b2, (short)0, acc2, false, false);
        acc3 = __builtin_amdgcn_wmma_f32_16x16x4_f32(false, a, false, b3, (short)0, acc3, false, false);
    }

    // Bias per output column n = nt*16 + l16 (same for both half-waves).
    const float bb0 = bias[0 * 16 + l16];
    const float bb1 = bias[1 * 16 + l16];
    const float bb2 = bias[2 * 16 + l16];
    const float bb3 = bias[3 * 16 + l16];

    // C/D layout: VGPR r, lane -> element (M = r + 8*half, N = l16)
#pragma unroll
    for (int r = 0; r < 8; ++r) {
        const int gr = m_tile * 16 + r + 8 * half;
        if (gr < n_nodes) {
            float* orow = out + (size_t)gr * OUT_DIM;
            orow[0 * 16 + l16] = acc0[r] + bb0;
            orow[1 * 16 + l16] = acc1[r] + bb1;
            orow[2 * 16 + l16] = acc2[r] + bb2;
            orow[3 * 16 + l16] = acc3[r] + bb3;
        }
    }
}

extern "C" void kernel_launch(void* const* d_in, const int* in_sizes, int n_in,
                              void* d_out, int out_size, void* d_ws, size_t ws_size,
                              hipStream_t stream) {
    const float* feat = (const float*)d_in[0];
    const int*   src  = (const int*)d_in[1];
    const int*   dst  = (const int*)d_in[2];
    const float* W    = (const float*)d_in[3];
    const float* bias = (const float*)d_in[4];
    float* out = (float*)d_out;
    float* agg = (float*)d_ws;   // [n_nodes * 64] fp32 scratch

    const int n_nodes = in_sizes[0] / IN_DIM;
    const int n_edges = in_sizes[1];
    const int total   = n_nodes * IN_DIM;

    // Phase 0: zero agg
    {
        int n4 = total / 4;
        int blocks = (n4 + 255) / 256;
        zero_kernel<<<blocks, 256, 0, stream>>>((float4*)agg, n4);
    }
    // Phase 1: scatter-add messages (1 wave32 per edge, 8 edges/block)
    {
        int blocks = (n_edges + 7) / 8;
        scatter_kernel<<<blocks, 256, 0, stream>>>(feat, src, dst, agg, n_edges);
    }
    // Phase 2: WMMA GEMM + bias
    {
        int m_tiles = (n_nodes + 15) / 16;
        int blocks = (m_tiles + 7) / 8;
        gemm_kernel<<<blocks, 256, 0, stream>>>(agg, W, bias, out, n_nodes);
    }
}